// GraphGAT_36249523978329
// MI455X (gfx1250) — compile-verified
//
#include <hip/hip_runtime.h>
#include <math.h>

#define N_NODES 50000
#define N_EDGES 800000
#define IN_FEATS 64
#define HD 64      // NUM_HEADS * N_HIDDEN
#define NH 4
#define DH 16

typedef __attribute__((ext_vector_type(2))) float v2f;
typedef __attribute__((ext_vector_type(8))) float v8f;

// ---------- order-preserving float<->int keys for atomicMax-based segment max ----------
__device__ __forceinline__ int f2ord(float f) {
  int i = __float_as_int(f);
  return i >= 0 ? i : (i ^ 0x7FFFFFFF);
}
__device__ __forceinline__ float ord2f(int i) {
  return __int_as_float(i >= 0 ? i : (i ^ 0x7FFFFFFF));
}

// ---------- WMMA f32 GEMM: out[n][o] = sum_k in[n][k] * W[o][k] ----------
// in: [N,64] row-major, W: [64,64] row-major (rows = output features).
// One wave: 16-node x 16-output tile, K marched 4 at a time via V_WMMA_F32_16X16X4_F32.
// 4 waves/block cover the 64 outputs; grid = N/16 blocks (N divisible by 16).
__global__ __launch_bounds__(128) void gat_gemm_wmma(const float* __restrict__ in,
                                                     const float* __restrict__ W,
                                                     float* __restrict__ out) {
  const int lane = threadIdx.x & 31;
  const int wave = threadIdx.x >> 5;     // N-tile 0..3
  const int half = lane >> 4;            // lane group
  const int r    = lane & 15;
  const int row  = blockIdx.x * 16 + r;  // node index (A: M = lane%16)
  const int col  = wave * 16 + r;        // output feature (B: N = lane%16)
  const float* __restrict__ arow = in + (size_t)row * IN_FEATS;
  const float* __restrict__ brow = W  + (size_t)col * IN_FEATS;

  v8f acc = {};
#pragma unroll
  for (int k0 = 0; k0 < IN_FEATS; k0 += 4) {
    // A-reg j holds K = k0 + 2*half + j  -> contiguous float2 per lane
    v2f a = *(const v2f*)(arow + k0 + 2 * half);
    v2f b = *(const v2f*)(brow + k0 + 2 * half);
    acc = __builtin_amdgcn_wmma_f32_16x16x4_f32(
        /*neg_a=*/false, a, /*neg_b=*/false, b,
        /*c_mod=*/(short)0, acc, /*reuse_a=*/false, /*reuse_b=*/false);
  }

  // C/D layout: VGPR i -> M = i + 8*half, N = lane%16
#pragma unroll
  for (int i = 0; i < 8; ++i) {
    int m = i + 8 * half;
    out[(size_t)(blockIdx.x * 16 + m) * HD + wave * 16 + r] = acc[i];
  }
}

// ---------- init accumulators for one layer ----------
__global__ void gat_init(float* __restrict__ out, int* __restrict__ mkey,
                         float* __restrict__ z) {
  int i = blockIdx.x * blockDim.x + threadIdx.x;
  if (i < N_NODES * HD) out[i] = 0.0f;
  if (i < N_NODES * NH) { mkey[i] = (int)0x80000000; z[i] = 0.0f; }
}

// ---------- per-edge attention scores + segment max over dst ----------
__global__ void gat_scores(const float* __restrict__ ft, const int* __restrict__ src,
                           const int* __restrict__ dst, float* __restrict__ esc,
                           int* __restrict__ mkey) {
  int e = blockIdx.x * blockDim.x + threadIdx.x;
  if (e >= N_EDGES) return;
  int s = src[e], d = dst[e];
  const float4* fs = (const float4*)(ft + (size_t)s * HD);
  const float4* fd = (const float4*)(ft + (size_t)d * HD);
#pragma unroll
  for (int h = 0; h < NH; ++h) {
    float acc = 0.0f;
#pragma unroll
    for (int q = 0; q < 4; ++q) {
      float4 a = fs[h * 4 + q];
      float4 b = fd[h * 4 + q];
      acc += a.x * b.x + a.y * b.y + a.z * b.z + a.w * b.w;
    }
    float sc = acc * 0.25f;  // 1/sqrt(16)
    esc[(size_t)e * NH + h] = sc;
    atomicMax(&mkey[d * NH + h], f2ord(sc));
  }
}

// ---------- exp(e - m[dst]) in place + segment sum z ----------
__global__ void gat_expz(float* __restrict__ esc, const int* __restrict__ dst,
                         const int* __restrict__ mkey, float* __restrict__ z) {
  int i = blockIdx.x * blockDim.x + threadIdx.x;
  if (i >= N_EDGES * NH) return;
  int e = i >> 2, h = i & 3;
  int d = dst[e];
  float m = ord2f(mkey[d * NH + h]);
  float ex = expf(esc[i] - m);
  esc[i] = ex;
  atomicAdd(&z[d * NH + h], ex);
}

// ---------- weighted scatter-sum: out[dst] += (ex/z[dst]) * ft[src] ----------
__global__ void gat_agg(const float* __restrict__ ft, const float* __restrict__ ex,
                        const float* __restrict__ z, const int* __restrict__ src,
                        const int* __restrict__ dst, float* __restrict__ out) {
  int i = blockIdx.x * blockDim.x + threadIdx.x;
  if (i >= N_EDGES * NH) return;
  int e = i >> 2, h = i & 3;
  int s = src[e], d = dst[e];
  float a = ex[i] / z[d * NH + h];
  const float4* fs = (const float4*)(ft + (size_t)s * HD + h * DH);
  float* o = out + (size_t)d * HD + h * DH;
#pragma unroll
  for (int q = 0; q < 4; ++q) {
    float4 v = fs[q];
    atomicAdd(o + q * 4 + 0, a * v.x);
    atomicAdd(o + q * 4 + 1, a * v.y);
    atomicAdd(o + q * 4 + 2, a * v.z);
    atomicAdd(o + q * 4 + 3, a * v.w);
  }
}

// ---------- ELU in place ----------
__global__ void gat_elu(float* __restrict__ h, int n) {
  int i = blockIdx.x * blockDim.x + threadIdx.x;
  if (i < n) {
    float x = h[i];
    h[i] = x > 0.0f ? x : expm1f(x);
  }
}

extern "C" void kernel_launch(void* const* d_in, const int* in_sizes, int n_in,
                              void* d_out, int out_size, void* d_ws, size_t ws_size,
                              hipStream_t stream) {
  const float* feat = (const float*)d_in[0];
  const int*   esrc = (const int*)d_in[1];
  const int*   edst = (const int*)d_in[2];
  const float* W1   = (const float*)d_in[3];
  const float* W2   = (const float*)d_in[4];
  float* out = (float*)d_out;

  float* ws   = (float*)d_ws;
  float* ft   = ws;                                   // N*64
  float* h1   = ft  + (size_t)N_NODES * HD;           // N*64
  float* esc  = h1  + (size_t)N_NODES * HD;           // E*4 (scores, then exp in place)
  int*   mkey = (int*)(esc + (size_t)N_EDGES * NH);   // N*4
  float* z    = (float*)(mkey + (size_t)N_NODES * NH);// N*4

  const int TB = 256;
  const int gInit = (N_NODES * HD + TB - 1) / TB;
  const int gE    = (N_EDGES + TB - 1) / TB;
  const int gEH   = (N_EDGES * NH + TB - 1) / TB;

  auto layer = [&](const float* fin, const float* W, float* oout) {
    gat_gemm_wmma<<<N_NODES / 16, 128, 0, stream>>>(fin, W, ft);
    gat_init<<<gInit, TB, 0, stream>>>(oout, mkey, z);
    gat_scores<<<gE, TB, 0, stream>>>(ft, esrc, edst, esc, mkey);
    gat_expz<<<gEH, TB, 0, stream>>>(esc, edst, mkey, z);
    gat_agg<<<gEH, TB, 0, stream>>>(ft, esc, z, esrc, edst, oout);
  };

  layer(feat, W1, h1);
  gat_elu<<<gInit, TB, 0, stream>>>(h1, N_NODES * HD);
  layer(h1, W2, out);
}